// SE_SORT_6408091205886
// MI455X (gfx1250) — compile-verified
//
#include <hip/hip_runtime.h>
#include <math.h>

typedef __attribute__((ext_vector_type(2))) float v2f;
typedef __attribute__((ext_vector_type(8))) float v8f;

#define NB  32      // batch
#define C1  512     // channels in
#define C2  256     // channels kept
#define CR  32      // C1 / RATIO (hidden dim)
#define HW  4096    // 64*64

// ---------------------------------------------------------------------------
// Kernel 1: global average pool. y[b*C1+c] = mean(x[b][c][:][:]).
// One block per (b,c): 256 threads * 4 float4 = 4096 floats, LDS reduce.
// ---------------------------------------------------------------------------
__global__ __launch_bounds__(256) void se_pool(const float* __restrict__ x,
                                               float* __restrict__ y) {
    __shared__ float red[256];
    const int bc = blockIdx.x;                       // b*C1 + c
    const float4* src = (const float4*)(x + (size_t)bc * HW);
    const int t = threadIdx.x;
    float acc = 0.f;
#pragma unroll
    for (int i = 0; i < 4; ++i) {
        float4 v = src[t + i * 256];
        acc += v.x + v.y + v.z + v.w;
    }
    red[t] = acc;
    __syncthreads();
    for (int off = 128; off > 0; off >>= 1) {
        if (t < off) red[t] += red[t + off];
        __syncthreads();
    }
    if (t == 0) y[bc] = red[0] * (1.0f / 4096.0f);
}

// ---------------------------------------------------------------------------
// Kernel 2: SE MLP entirely with V_WMMA_F32_16X16X4_F32 (exact f32 math).
//   H = relu(Y[32x512] @ W1^T[512x32])   (w1 is [32,512] row-major)
//   S = sigmoid(H[32x32] @ W2^T[32x512]) (w2 is [512,32] row-major)
// Single block, 4 waves. Per-lane operand packing per ISA 7.12.2 (f32):
//   A 16x4 : lanes 0-15 -> M=lane, VGPR{0,1} = K{0,1}; lanes 16-31 -> K{2,3}
//   B 4x16 : mirrored (lanes 0-15 -> N=lane, K rows {0,1}; hi lanes K{2,3})
//   C/D    : VGPR r -> row r (lanes 0-15) / row r+8 (lanes 16-31), N=lane%16
// ---------------------------------------------------------------------------
__global__ __launch_bounds__(128) void se_mlp(const float* __restrict__ y,
                                              const float* __restrict__ w1,
                                              const float* __restrict__ w2,
                                              float* __restrict__ s) {
    __shared__ float H[32][33];                      // pad to avoid bank conflicts
    const int tid  = threadIdx.x;
    const int wave = tid >> 5;
    const int lane = tid & 31;
    const int lr   = lane & 15;
    const int hi   = lane >> 4;                      // 0 or 1
    const int kl   = hi << 1;                        // K sub-offset: 0 or 2

    // ---- GEMM1: one 16x16 tile of H per wave, K = 512 (128 WMMAs) ----
    {
        const int ti = wave >> 1, tj = wave & 1;
        const float* ya = y  + (size_t)(ti * 16 + lr) * C1;   // A row = batch
        const float* wb = w1 + (size_t)(tj * 16 + lr) * C1;   // B col = w1 row
        v8f acc = {};
        for (int k0 = 0; k0 < C1; k0 += 4) {
            v2f a, b;
            a.x = ya[k0 + kl]; a.y = ya[k0 + kl + 1];
            b.x = wb[k0 + kl]; b.y = wb[k0 + kl + 1];
            acc = __builtin_amdgcn_wmma_f32_16x16x4_f32(
                false, a, false, b, (short)0, acc, false, false);
        }
#pragma unroll
        for (int r = 0; r < 8; ++r) {
            float v = acc[r];
            H[ti * 16 + r + hi * 8][tj * 16 + lr] = v > 0.f ? v : 0.f; // ReLU
        }
    }
    __syncthreads();

    // ---- GEMM2: 64 output tiles (2 row x 32 col), 16 per wave, K = 32 ----
    for (int t = wave; t < 64; t += 4) {
        const int ti = t & 1;                        // batch-row tile
        const int tc = t >> 1;                       // channel-col tile
        const float* wb = w2 + (size_t)(tc * 16 + lr) * CR;   // B col = w2 row
        v8f acc = {};
#pragma unroll
        for (int k0 = 0; k0 < CR; k0 += 4) {
            v2f a, b;
            a.x = H[ti * 16 + lr][k0 + kl]; a.y = H[ti * 16 + lr][k0 + kl + 1];
            b.x = wb[k0 + kl];              b.y = wb[k0 + kl + 1];
            acc = __builtin_amdgcn_wmma_f32_16x16x4_f32(
                false, a, false, b, (short)0, acc, false, false);
        }
#pragma unroll
        for (int r = 0; r < 8; ++r) {
            float v = acc[r];
            s[(size_t)(ti * 16 + r + hi * 8) * C1 + tc * 16 + lr] =
                1.0f / (1.0f + expf(-v));            // sigmoid
        }
    }
}

// ---------------------------------------------------------------------------
// Kernel 3: exact stable argsort-descending via rank-by-count.
// rank(c) = #{j : s[j] > s[c]}  +  #{j < c : s[j] == s[c]}
// One block per batch, thread c computes its own rank; scatter if rank < C2.
// ---------------------------------------------------------------------------
__global__ __launch_bounds__(512) void se_rank(const float* __restrict__ s,
                                               int* __restrict__ idx) {
    __shared__ float sl[C1];
    const int b = blockIdx.x;
    const int c = threadIdx.x;
    sl[c] = s[(size_t)b * C1 + c];
    __syncthreads();
    const float v = sl[c];
    int rank = 0;
    for (int j = 0; j < C1; ++j) {
        const float sj = sl[j];
        rank += (sj > v) || (sj == v && j < c);
    }
    if (rank < C2) idx[b * C2 + rank] = c;
}

// ---------------------------------------------------------------------------
// Kernel 4: gather selected channels. One block per (b, j): copy 16 KB
// contiguous channel plane with float4 (global_load_b128 / store_b128).
// ---------------------------------------------------------------------------
__global__ __launch_bounds__(256) void se_gather(const float* __restrict__ x,
                                                 const int* __restrict__ idx,
                                                 float* __restrict__ out) {
    const int bj = blockIdx.x;                       // b*C2 + j
    const int b  = bj >> 8;                          // / C2
    const int c  = idx[bj];                          // selected channel
    const float4* src = (const float4*)(x + ((size_t)b * C1 + c) * HW);
    float4*       dst = (float4*)(out + (size_t)bj * HW);
    const int t = threadIdx.x;
#pragma unroll
    for (int i = 0; i < 4; ++i) dst[t + i * 256] = src[t + i * 256];
}

// ---------------------------------------------------------------------------
extern "C" void kernel_launch(void* const* d_in, const int* in_sizes, int n_in,
                              void* d_out, int out_size, void* d_ws, size_t ws_size,
                              hipStream_t stream) {
    (void)in_sizes; (void)n_in; (void)out_size; (void)ws_size;
    const float* x  = (const float*)d_in[0];   // [32,512,64,64]
    const float* w1 = (const float*)d_in[1];   // [32,512]
    const float* w2 = (const float*)d_in[2];   // [512,32]
    float* out = (float*)d_out;                // [32,256,64,64]

    // workspace layout: y[32*512] f32 | s[32*512] f32 | idx[32*256] i32
    float* y   = (float*)d_ws;
    float* s   = y + NB * C1;
    int*   idx = (int*)(s + NB * C1);

    se_pool  <<<dim3(NB * C1), 256, 0, stream>>>(x, y);
    se_mlp   <<<dim3(1),       128, 0, stream>>>(y, w1, w2, s);
    se_rank  <<<dim3(NB),      512, 0, stream>>>(s, idx);
    se_gather<<<dim3(NB * C2), 256, 0, stream>>>(x, idx, out);
}